// GCN_Inv_Phys2_50096498541183
// MI455X (gfx1250) — compile-verified
//
#include <hip/hip_runtime.h>
#include <math.h>

typedef float v2f __attribute__((ext_vector_type(2)));
typedef float v8f __attribute__((ext_vector_type(8)));

#define LW 128        // hidden width
#define ROWS 64       // rows per block (4 sub-tiles of 16)
#define LDST 260      // LDS row stride (floats): 260 % 64 == 4 -> conflict-free column reads

// ---------------- utility kernels ----------------

__global__ void zero_kernel(float4* __restrict__ p, long n4) {
  long i = (long)blockIdx.x * blockDim.x + threadIdx.x;
  long stride = (long)gridDim.x * blockDim.x;
  const float4 z = make_float4(0.f, 0.f, 0.f, 0.f);
  for (; i < n4; i += stride) p[i] = z;
}

__global__ void deg_kernel(const int* __restrict__ ei, float* __restrict__ deg, int E) {
  int e = blockIdx.x * blockDim.x + threadIdx.x;
  if (e < E) {
    atomicAdd(&deg[ei[2 * e + 0]], 1.0f);
    atomicAdd(&deg[ei[2 * e + 1]], 1.0f);
  }
}

__global__ void rdeg_kernel(float* __restrict__ deg, int N) {
  int n = blockIdx.x * blockDim.x + threadIdx.x;
  if (n < N) deg[n] = 1.0f / fmaxf(deg[n], 1.0f);
}

__global__ void featT_kernel(const float* __restrict__ fn, const float* __restrict__ T,
                             float* __restrict__ featT, int N) {
  int n = blockIdx.x * blockDim.x + threadIdx.x;
  if (n >= N) return;
  #pragma unroll
  for (int j = 0; j < 9; ++j) featT[(long)n * 10 + j] = fn[(long)n * 9 + j];
  featT[(long)n * 10 + 9] = T[n];
}

// ---------------- WMMA GEMM kernels ----------------
// One block: 64 rows x 128 cols, 8 waves. Each wave owns one 16-col stripe and
// 4 independent 16x16 fp32 WMMA accumulators (4 row sub-tiles), reusing each
// B float2 across 4 back-to-back v_wmma_f32_16x16x4_f32.
// A (64 x Kpad) staged zero-padded in LDS; B read from global (L2-resident);
// K-tail handled by clamping the B address (A zero-padding kills the term),
// so the inner loop has no divergent branches.

__global__ __launch_bounds__(256) void edge_gcn_kernel(
    const float* __restrict__ fgp, int Kg,
    const float* __restrict__ fnode, int Kn,
    const int* __restrict__ ei,
    const float* __restrict__ W, const float* __restrict__ bias,
    float* __restrict__ hgp_out, float* __restrict__ agg, int E)
{
  const int K = Kg + Kn;
  const int Kpad = (K + 3) & ~3;
  __shared__ float As[ROWS][LDST];

  const int t = threadIdx.x;
  const long e0 = (long)blockIdx.x * ROWS;

  for (int rr = (t >> 4); rr < ROWS; rr += 16) {
    long e = e0 + rr; if (e >= E) e = E - 1;        // clamp: EXEC stays all-ones
    const int s = ei[2 * e + 0];
    const int d = ei[2 * e + 1];
    for (int k = (t & 15); k < Kpad; k += 16) {
      float v = 0.0f;
      if (k < Kg)      v = fgp[e * Kg + k];
      else if (k < K)  { int j = k - Kg;
                         v = 0.5f * (fnode[(long)s * Kn + j] + fnode[(long)d * Kn + j]); }
      As[rr][k] = v;
    }
  }
  __syncthreads();

  const int w    = t >> 5;        // wave 0..7 -> 16-col stripe
  const int lane = t & 31;
  const int half = lane >> 4;
  const int m    = lane & 15;
  const int col  = w * 16 + m;

  v8f acc[4];
  const float b0 = bias[col];
  #pragma unroll
  for (int s2 = 0; s2 < 4; ++s2)
    #pragma unroll
    for (int v = 0; v < 8; ++v) acc[s2][v] = b0;

  const int K4 = K & ~3;
  for (int k0 = 0; k0 < K4; k0 += 4) {             // unguarded hot loop
    const int kk = k0 + 2 * half;
    v2f b; b.x = W[(long)kk * LW + col];
           b.y = W[(long)(kk + 1) * LW + col];
    #pragma unroll
    for (int s2 = 0; s2 < 4; ++s2) {
      v2f a; a.x = As[s2 * 16 + m][kk]; a.y = As[s2 * 16 + m][kk + 1];
      acc[s2] = __builtin_amdgcn_wmma_f32_16x16x4_f32(false, a, false, b, (short)0,
                                                      acc[s2], false, false);
    }
  }
  if (K4 < Kpad) {                                  // single branchless tail step
    const int kk = K4 + 2 * half;
    const int kx = (kk     < K) ? kk     : K - 1;   // clamp addr; A pad is zero
    const int ky = (kk + 1 < K) ? kk + 1 : K - 1;
    v2f b; b.x = W[(long)kx * LW + col];
           b.y = W[(long)ky * LW + col];
    #pragma unroll
    for (int s2 = 0; s2 < 4; ++s2) {
      v2f a; a.x = As[s2 * 16 + m][kk]; a.y = As[s2 * 16 + m][kk + 1];
      acc[s2] = __builtin_amdgcn_wmma_f32_16x16x4_f32(false, a, false, b, (short)0,
                                                      acc[s2], false, false);
    }
  }

  #pragma unroll
  for (int s2 = 0; s2 < 4; ++s2) {
    #pragma unroll
    for (int v = 0; v < 8; ++v) {
      const int row = s2 * 16 + v + 8 * half;
      const long e = e0 + row;
      if (e < E) {
        const float val = fmaxf(acc[s2][v], 0.0f);  // ReLU
        if (hgp_out) hgp_out[e * LW + col] = val;
        const int s = ei[2 * e + 0];
        const int d = ei[2 * e + 1];
        atomicAdd(&agg[(long)s * LW + col], val);
        atomicAdd(&agg[(long)d * LW + col], val);
      }
    }
  }
}

__global__ __launch_bounds__(256) void node_gcn_kernel(
    const float* __restrict__ fnode, int Kn,
    const float* __restrict__ agg, const float* __restrict__ rdeg,
    const float* __restrict__ W, const float* __restrict__ bias,
    float* __restrict__ hnode_out, int N)
{
  const int K = Kn + LW;
  const int Kpad = (K + 3) & ~3;
  __shared__ float As[ROWS][LDST];

  const int t = threadIdx.x;
  const long n0 = (long)blockIdx.x * ROWS;

  for (int rr = (t >> 4); rr < ROWS; rr += 16) {
    long n = n0 + rr; if (n >= N) n = N - 1;
    const float rd = rdeg[n];
    for (int k = (t & 15); k < Kpad; k += 16) {
      float v = 0.0f;
      if (k < Kn)     v = fnode[n * Kn + k];
      else if (k < K) v = agg[n * LW + (k - Kn)] * rd;   // scatter-mean
      As[rr][k] = v;
    }
  }
  __syncthreads();

  const int w    = t >> 5;
  const int lane = t & 31;
  const int half = lane >> 4;
  const int m    = lane & 15;
  const int col  = w * 16 + m;

  v8f acc[4];
  const float b0 = bias[col];
  #pragma unroll
  for (int s2 = 0; s2 < 4; ++s2)
    #pragma unroll
    for (int v = 0; v < 8; ++v) acc[s2][v] = b0;

  const int K4 = K & ~3;
  for (int k0 = 0; k0 < K4; k0 += 4) {
    const int kk = k0 + 2 * half;
    v2f b; b.x = W[(long)kk * LW + col];
           b.y = W[(long)(kk + 1) * LW + col];
    #pragma unroll
    for (int s2 = 0; s2 < 4; ++s2) {
      v2f a; a.x = As[s2 * 16 + m][kk]; a.y = As[s2 * 16 + m][kk + 1];
      acc[s2] = __builtin_amdgcn_wmma_f32_16x16x4_f32(false, a, false, b, (short)0,
                                                      acc[s2], false, false);
    }
  }
  if (K4 < Kpad) {
    const int kk = K4 + 2 * half;
    const int kx = (kk     < K) ? kk     : K - 1;
    const int ky = (kk + 1 < K) ? kk + 1 : K - 1;
    v2f b; b.x = W[(long)kx * LW + col];
           b.y = W[(long)ky * LW + col];
    #pragma unroll
    for (int s2 = 0; s2 < 4; ++s2) {
      v2f a; a.x = As[s2 * 16 + m][kk]; a.y = As[s2 * 16 + m][kk + 1];
      acc[s2] = __builtin_amdgcn_wmma_f32_16x16x4_f32(false, a, false, b, (short)0,
                                                      acc[s2], false, false);
    }
  }

  #pragma unroll
  for (int s2 = 0; s2 < 4; ++s2) {
    #pragma unroll
    for (int v = 0; v < 8; ++v) {
      const long n = n0 + s2 * 16 + v + 8 * half;
      if (n < N) hnode_out[n * LW + col] = fmaxf(acc[s2][v], 0.0f);
    }
  }
}

// ---------------- small heads + combine ----------------

__global__ void head_kernel(const float* __restrict__ h, const float* __restrict__ W,
                            const float* __restrict__ b, float* __restrict__ out,
                            int N, int C) {
  int n = blockIdx.x * blockDim.x + threadIdx.x;
  if (n >= N) return;
  const float4* hp = (const float4*)(h + (long)n * LW);
  for (int cc = 0; cc < C; ++cc) {
    float s = b[cc];
    #pragma unroll 8
    for (int k = 0; k < LW / 4; ++k) {
      float4 hv = hp[k];
      s += hv.x * W[(4 * k + 0) * C + cc] + hv.y * W[(4 * k + 1) * C + cc] +
           hv.z * W[(4 * k + 2) * C + cc] + hv.w * W[(4 * k + 3) * C + cc];
    }
    out[(long)n * C + cc] = s;
  }
}

__global__ void combine_kernel(const float* __restrict__ coeff, const float* __restrict__ eh,
                               const float* __restrict__ bh, const float* __restrict__ sh,
                               const float* __restrict__ tanf, const float* __restrict__ T,
                               float* __restrict__ out, int N) {
  int n = blockIdx.x * blockDim.x + threadIdx.x;
  if (n >= N) return;
  const float c0 = coeff[2 * n + 0], c1 = coeff[2 * n + 1];
  const float Eg = expf(-fabsf(eh[n]) / T[n]);
  const float Bv = fabsf(bh[n]);
  #pragma unroll
  for (int j = 0; j < 3; ++j) {
    const float o = c0 * tanf[(long)n * 6 + j] + c1 * tanf[(long)n * 6 + 3 + j];
    out[(long)n * 3 + j] = o / Bv * Eg;
  }
  out[(long)3 * N + n] = fabsf(sh[n]);
}

// ---------------- orchestration ----------------

extern "C" void kernel_launch(void* const* d_in, const int* in_sizes, int n_in,
                              void* d_out, int out_size, void* d_ws, size_t ws_size,
                              hipStream_t stream) {
  const int E = in_sizes[0] / 2;     // edge_index (E,2)
  const int N = in_sizes[5];         // feature_T (N,1)

  const int*   ei    = (const int*)d_in[0];
  const float* fGP   = (const float*)d_in[1];
  const float* fNode = (const float*)d_in[2];
  const float* fBd   = (const float*)d_in[3];
  const float* fTan  = (const float*)d_in[4];
  const float* fT    = (const float*)d_in[5];
  // params flattened in insertion order: conv1{Wg,bg,Wn,bn}, conv2{...}, linear_W, linear_b,
  // Bconv1, Bconv2, Blinear_W, Blinear_b, Econv1, Econv2, Elinear_W, Elinear_b,
  // sigconv1, sigconv2, siglinear_W, siglinear_b  -> tower bases 6, 16, 26, 36.
  auto P = [&](int i) { return (const float*)d_in[i]; };

  float* ws = (float*)d_ws;
  size_t off = 0;
  auto alloc = [&](size_t nf) { float* p = ws + off; off += (nf + 63) & ~(size_t)63; return p; };
  float* rdeg  = alloc(N);
  float* agg   = alloc((size_t)N * LW);
  float* hgp   = alloc((size_t)E * LW);
  float* hn1   = alloc((size_t)N * LW);
  float* hn2   = alloc((size_t)N * LW);
  float* coeff = alloc((size_t)N * 2);
  float* eh    = alloc(N);
  float* bh    = alloc(N);
  float* sh    = alloc(N);
  float* featT = alloc((size_t)N * 10);

  const int tb = 256;
  const int eb = (E + ROWS - 1) / ROWS;
  const int nb = (N + ROWS - 1) / ROWS;

  // degrees (shared by all 8 GCN calls) + featT
  zero_kernel<<<256, tb, 0, stream>>>((float4*)rdeg, N / 4);
  deg_kernel<<<(E + tb - 1) / tb, tb, 0, stream>>>(ei, rdeg, E);
  rdeg_kernel<<<(N + tb - 1) / tb, tb, 0, stream>>>(rdeg, N);
  featT_kernel<<<(N + tb - 1) / tb, tb, 0, stream>>>(fNode, fT, featT, N);

  auto tower = [&](const float* fgp, int Kg, const float* fnode, int Kn, int p,
                   float* headout, int C, int hwIdx, int hbIdx) {
    zero_kernel<<<2048, tb, 0, stream>>>((float4*)agg, (long)N * LW / 4);
    edge_gcn_kernel<<<eb, tb, 0, stream>>>(fgp, Kg, fnode, Kn, ei, P(p), P(p + 1), hgp, agg, E);
    node_gcn_kernel<<<nb, tb, 0, stream>>>(fnode, Kn, agg, rdeg, P(p + 2), P(p + 3), hn1, N);
    zero_kernel<<<2048, tb, 0, stream>>>((float4*)agg, (long)N * LW / 4);
    edge_gcn_kernel<<<eb, tb, 0, stream>>>(hgp, LW, hn1, LW, ei, P(p + 4), P(p + 5), nullptr, agg, E);
    node_gcn_kernel<<<nb, tb, 0, stream>>>(hn1, LW, agg, rdeg, P(p + 6), P(p + 7), hn2, N);
    head_kernel<<<(N + tb - 1) / tb, tb, 0, stream>>>(hn2, P(hwIdx), P(hbIdx), headout, N, C);
  };

  tower(fGP, 6, fNode, 9,  6,  coeff, 2, 14, 15);  // main -> coeff
  tower(fBd, 1, featT, 10, 16, bh,    1, 24, 25);  // B    -> |b head|
  tower(fGP, 6, fNode, 9,  26, eh,    1, 34, 35);  // E    -> exp(-|.|/T)
  tower(fGP, 6, featT, 10, 36, sh,    1, 44, 45);  // sig  -> |.|

  combine_kernel<<<(N + tb - 1) / tb, tb, 0, stream>>>(coeff, eh, bh, sh, fTan, fT,
                                                       (float*)d_out, N);
}